// TopKLayer_16698832847319
// MI455X (gfx1250) — compile-verified
//
#include <hip/hip_runtime.h>
#include <stdint.h>

// Problem constants (from reference): x is (64, 256, 56, 56) fp32.
#define HW     3136        // 56*56
#define KSEL   313         // int(max(1, 0.1*3136))
#define NROWS  (64 * 256)  // independent rows
#define NT     256         // threads per block = 8 waves (wave32)
#define NW     (NT / 32)   // 8 waves
#define CH     13          // ceil(HW / NT): contiguous elements per thread

// Exact pointee type the async builtins want (verified round 1 diagnostic):
typedef int v4i __attribute__((vector_size(4 * sizeof(int))));
typedef __attribute__((address_space(1))) v4i GV4;  // global int4
typedef __attribute__((address_space(3))) v4i LV4;  // LDS int4

__device__ __forceinline__ void async_wait0() {
#if __has_builtin(__builtin_amdgcn_s_wait_asynccnt)
    __builtin_amdgcn_s_wait_asynccnt(0);
#else
    asm volatile("s_wait_asynccnt 0" ::: "memory");
#endif
}

__global__ __launch_bounds__(NT) void topk_rows_kernel(const float* __restrict__ x,
                                                       float* __restrict__ out) {
    __shared__ __align__(16) float srow[HW];    // one full row (12544 B)
    __shared__ uint32_t histw[NW][256];         // per-wave privatized histograms (8 KB)
    __shared__ uint32_t wsum[NW];               // cross-wave prefix fixup
    __shared__ uint32_t s_bin, s_above;

    const int tid  = threadIdx.x;
    const int w    = tid >> 5;
    const int lane = tid & 31;
    const int row  = blockIdx.x;
    const float* gin  = x   + (size_t)row * HW;  // 12544-byte row stride -> 16B aligned
    float*       gout = out + (size_t)row * HW;

    // ---------- Stage row: global -> LDS via async DMA (ASYNCcnt path) ----------
    const int NV = HW / 4;  // 784 b128 tiles
    for (int v = tid; v < NV; v += NT) {
#if __has_builtin(__builtin_amdgcn_global_load_async_to_lds_b128)
        __builtin_amdgcn_global_load_async_to_lds_b128(
            (GV4*)(gin + 4 * v), (LV4*)(&srow[4 * v]), 0, 0);
#else
        ((float4*)srow)[v] = ((const float4*)gin)[v];
#endif
    }
    async_wait0();
    __syncthreads();

    // ---------- Cache my contiguous 13-element chunk in registers ----------
    const int beg = tid * CH;
    float f[CH];
#pragma unroll
    for (int j = 0; j < CH; ++j) {
        const int i = beg + j;
        f[j] = (i < HW) ? srow[i] : 0.0f;  // pad value never counted (guarded below)
    }

    // ---------- 4-pass radix select (8 bits/pass) on key = bits(|x|) ----------
    // Invariant after each pass: remaining = KSEL - count(key > current prefix path)
    uint32_t prefix    = 0;
    uint32_t remaining = KSEL;
    for (int pass = 0; pass < 4; ++pass) {
        const int shift = 24 - 8 * pass;

        // Zero the 8 private histograms (2048 entries / 256 threads = 8 each)
#pragma unroll
        for (int q = 0; q < NW; ++q) ((uint32_t*)histw)[tid + q * NT] = 0u;
        __syncthreads();

        // Count into my wave's private histogram (8x less atomic contention)
#pragma unroll
        for (int j = 0; j < CH; ++j) {
            const int i = beg + j;
            if (i < HW) {
                const uint32_t u = __float_as_uint(f[j]) & 0x7fffffffu;
                const bool match = (pass == 0) || ((u >> (shift + 8)) == prefix);
                if (match) atomicAdd(&histw[w][(u >> shift) & 0xffu], 1u);  // ds_add_u32
            }
        }
        __syncthreads();

        // Wave 0 reduces the 8 copies + suffix-scans 256 bins with shuffles,
        // and picks the k-th bin directly from registers. 0 extra barriers.
        if (w == 0) {
            // lane owns bins [lane*8, lane*8+8)
            uint32_t b[8];
#pragma unroll
            for (int j = 0; j < 8; ++j) {
                uint32_t s = 0;
#pragma unroll
                for (int q = 0; q < NW; ++q) s += histw[q][lane * 8 + j];
                b[j] = s;
            }
            // local suffix sums within my 8 bins
            uint32_t ls[8];
            uint32_t acc = 0;
#pragma unroll
            for (int j = 7; j >= 0; --j) { acc += b[j]; ls[j] = acc; }
            const uint32_t mysum = acc;
            // inclusive suffix sum across lanes (higher lane = higher bins)
            uint32_t inc = mysum;
#pragma unroll
            for (int off = 1; off < 32; off <<= 1) {
                const uint32_t o = __shfl_down(inc, off, 32);
                if (lane + off < 32) inc += o;
            }
            const uint32_t excl = inc - mysum;  // sum over lanes > lane
            // select: largest bin b* with suffix(b*) >= remaining
#pragma unroll
            for (int j = 0; j < 8; ++j) {
                const uint32_t st  = ls[j] + excl;                       // suffix(bin)
                const uint32_t st1 = ((j < 7) ? ls[j + 1] : 0u) + excl;  // suffix(bin+1)
                if (st >= remaining && st1 < remaining) {  // exactly one (lane,j) hits
                    s_bin   = (uint32_t)(lane * 8 + j);
                    s_above = st1;
                }
            }
        }
        __syncthreads();
        prefix     = (prefix << 8) | s_bin;
        remaining -= s_above;
    }
    const uint32_t T   = prefix;     // k-th largest |x| bit pattern
    const uint32_t rem = remaining;  // how many elements == T to keep (lowest index first)

    // ---------- Tie handling: keep first `rem` equals in index order ----------
    uint32_t eq = 0;
#pragma unroll
    for (int j = 0; j < CH; ++j) {
        const int i = beg + j;
        if (i < HW) {
            const uint32_t u = __float_as_uint(f[j]) & 0x7fffffffu;
            eq += (u == T) ? 1u : 0u;
        }
    }
    // inclusive prefix within my wave via shuffles
    uint32_t incw = eq;
#pragma unroll
    for (int off = 1; off < 32; off <<= 1) {
        const uint32_t o = __shfl_up(incw, off, 32);
        if (lane >= off) incw += o;
    }
    if (lane == 31) wsum[w] = incw;
    __syncthreads();
    uint32_t wbase = 0;
#pragma unroll
    for (int q = 0; q < NW; ++q) wbase += (q < w) ? wsum[q] : 0u;
    const uint32_t excl   = wbase + incw - eq;                 // equals before my chunk
    uint32_t       budget = (excl < rem) ? (rem - excl) : 0u;  // equals I may keep

    // ---------- Mask in place in LDS ----------
#pragma unroll
    for (int j = 0; j < CH; ++j) {
        const int i = beg + j;
        if (i < HW) {
            const float    xv = f[j];
            const uint32_t u  = __float_as_uint(xv) & 0x7fffffffu;
            bool keep;
            if (u > T) {
                keep = true;
            } else if (u == T && budget > 0u) {
                keep = true;
                --budget;
            } else {
                keep = false;
            }
            srow[i] = keep ? xv : 0.0f;
        }
    }
    __syncthreads();

    // ---------- Stream row back: LDS -> global via async DMA ----------
    for (int v = tid; v < NV; v += NT) {
#if __has_builtin(__builtin_amdgcn_global_store_async_from_lds_b128)
        __builtin_amdgcn_global_store_async_from_lds_b128(
            (GV4*)(gout + 4 * v), (LV4*)(&srow[4 * v]), 0, 0);
#else
        ((float4*)gout)[v] = ((const float4*)srow)[v];
#endif
    }
    async_wait0();  // drain before end (s_endpgm would also wait-idle)
}

extern "C" void kernel_launch(void* const* d_in, const int* in_sizes, int n_in,
                              void* d_out, int out_size, void* d_ws, size_t ws_size,
                              hipStream_t stream) {
    const float* x   = (const float*)d_in[0];
    float*       out = (float*)d_out;
    topk_rows_kernel<<<NROWS, NT, 0, stream>>>(x, out);
}